// AP_38422777430169
// MI455X (gfx1250) — compile-verified
//
#include <hip/hip_runtime.h>
#include <hip/hip_bf16.h>
#include <cstdint>

// ---------------------------------------------------------------------------
// AP (average precision) for temporal detection.
// Pipeline:
//   k_masks : 256-bit IoU>0.5 mask per proposal (+ per-chunk OR, + compact conf)
//             labels staged into LDS via the CDNA5 Tensor Data Mover.
//   k_greedy: exact parallelization of the sequential greedy matcher
//             (single workgroup, min-index extraction, chunk skipping).
//   k_ranks : stable-sort rank of each TP confidence (compare-count).
//   k_ap    : suffix-max precision over TPs -> scalar AP.
// ---------------------------------------------------------------------------

#define N_PROPOSALS 100000
#define N_LABELS    256
#define CHUNK       1024
#define NCHUNK      ((N_PROPOSALS + CHUNK - 1) / CHUNK)   // 98

// d_ws layout (bytes)
#define OFF_MASK   0u            // 100000 * 8 u32  = 3,200,000
#define OFF_CONF   3200000u      // 100000 f32      =   400,000
#define OFF_COR    3600000u      // 98 * 8 u32      =     3,136
#define OFF_TPIDX  3603136u      // 256 i32
#define OFF_TPCONF 3604160u      // 256 f32
#define OFF_NTP    3605184u      // 1 i32
#define OFF_RANKS  3605248u      // 256 u32

typedef unsigned int v4u __attribute__((ext_vector_type(4)));
typedef int          v8i __attribute__((ext_vector_type(8)));
typedef int          v4i __attribute__((ext_vector_type(4)));

// ---------------------------------------------------------------------------
// Kernel 1: per-proposal 256-bit over-threshold masks + chunk OR + conf copy
// ---------------------------------------------------------------------------
__global__ __launch_bounds__(1024) void k_masks(const float* __restrict__ props,
                                                const float* __restrict__ labels,
                                                uint32_t* __restrict__ masks,
                                                float* __restrict__ conf,
                                                uint32_t* __restrict__ chunk_or) {
    __shared__ float    shl[2 * N_LABELS];   // 2 KB of label [start,end] pairs
    __shared__ uint32_t sor[8];

    const int tid = threadIdx.x;
    if (tid < 8) sor[tid] = 0u;

    // --- Tensor Data Mover: DMA 512 consecutive f32 labels -> LDS ----------
    if (tid == 0) {
        const uint64_t ga = (uint64_t)(uintptr_t)labels;
        const uint32_t la = (uint32_t)(uintptr_t)(&shl[0]);   // LDS byte offset
        v4u g0;
        g0.x = 1u;                                    // count=1 (valid D#)
        g0.y = la;                                    // lds_addr
        g0.z = (uint32_t)(ga & 0xFFFFFFFFull);        // global_addr[31:0]
        g0.w = (uint32_t)((ga >> 32) & 0x1FFFFFFull)  // global_addr[56:32]
             | (2u << 30);                            // type=2 ("image")
        v8i g1;
        g1[0] = (int)(2u << 16);     // data_size = 4 bytes; wg_mask=0 (no cluster)
        g1[1] = (int)(512u << 16);   // tensor_dim0 = 512 (low 16 in [31:16])
        g1[2] = (int)(1u << 16);     // tensor_dim0 hi=0 ; tensor_dim1 = 1
        g1[3] = (int)(512u << 16);   // tensor_dim1 hi=0 ; tile_dim0 = 512
        g1[4] = 1;                   // tile_dim1 = 1 ; tile_dim2 = 0
        g1[5] = 512;                 // tensor_dim0_stride[31:0] = 512
        g1[6] = 0;
        g1[7] = 0;
        v4i gz4 = {0, 0, 0, 0};          // groups 2/3 unused (<=2D tile)
        v8i gz8 = {0, 0, 0, 0, 0, 0, 0, 0};
        __builtin_amdgcn_tensor_load_to_lds(g0, g1, gz4, gz4, gz8, 0);
        __builtin_amdgcn_s_wait_tensorcnt(0);
    }
    __syncthreads();

    const int  i     = blockIdx.x * CHUNK + tid;
    const bool valid = (i < N_PROPOSALS);

    float amin = 0.f, amax = 0.f, la = 0.f, cf = 0.f;
    if (valid) {
        cf   = props[i * 3 + 0];
        amin = props[i * 3 + 1] / 25.0f;     // FPS division, matches reference
        amax = props[i * 3 + 2] / 25.0f;
        la   = amax - amin;
    }

    uint32_t m[8];
    #pragma unroll
    for (int w = 0; w < 8; ++w) {
        uint32_t bits = 0u;
        if (valid) {
            for (int b = 0; b < 32; ++b) {
                const int   j    = w * 32 + b;
                const float bmin = shl[2 * j];
                const float bmax = shl[2 * j + 1];
                float inter = fminf(amax, bmax) - fmaxf(amin, bmin);
                inter = fmaxf(inter, 0.0f);
                const float uni = la - inter + (bmax - bmin);
                const float iou = inter / uni;            // IEEE div, as reference
                if (iou > 0.5f) bits |= (1u << b);
            }
        }
        m[w] = bits;
    }

    if (valid) {
        uint4* mp = (uint4*)(masks + (size_t)i * 8);
        mp[0] = make_uint4(m[0], m[1], m[2], m[3]);
        mp[1] = make_uint4(m[4], m[5], m[6], m[7]);
        conf[i] = cf;
    }
    #pragma unroll
    for (int w = 0; w < 8; ++w)
        if (m[w]) atomicOr(&sor[w], m[w]);
    __syncthreads();
    if (tid < 8) chunk_or[blockIdx.x * 8 + tid] = sor[tid];
}

// ---------------------------------------------------------------------------
// Kernel 2: exact greedy matching (single workgroup, wave32-friendly)
// ---------------------------------------------------------------------------
__global__ __launch_bounds__(1024) void k_greedy(const uint32_t* __restrict__ masks,
                                                 const uint32_t* __restrict__ chunk_or,
                                                 const float* __restrict__ conf,
                                                 int* __restrict__ tp_idx,
                                                 float* __restrict__ tp_conf,
                                                 int* __restrict__ n_tp) {
    __shared__ int s_min;
    __shared__ int s_j;
    __shared__ int s_ntp;

    const int tid = threadIdx.x;
    if (tid == 0) s_ntp = 0;
    __syncthreads();

    uint32_t D[8];                    // 256-bit detected mask (replicated)
    #pragma unroll
    for (int w = 0; w < 8; ++w) D[w] = 0u;

    for (int c = 0; c < NCHUNK; ++c) {
        // Chunk-level skip: if no proposal in this chunk can match a free
        // label, the whole chunk is FP and D is unchanged across it.
        uint32_t anyw = 0u;
        #pragma unroll
        for (int w = 0; w < 8; ++w) anyw |= (chunk_or[c * 8 + w] & ~D[w]);
        if (anyw == 0u) continue;     // uniform: D identical on all threads

        const int i = c * CHUNK + tid;
        uint32_t m[8];
        if (i < N_PROPOSALS) {
            const uint4* mp = (const uint4*)(masks + (size_t)i * 8);
            const uint4 a = mp[0], b = mp[1];
            m[0] = a.x; m[1] = a.y; m[2] = a.z; m[3] = a.w;
            m[4] = b.x; m[5] = b.y; m[6] = b.z; m[7] = b.w;
        } else {
            #pragma unroll
            for (int w = 0; w < 8; ++w) m[w] = 0u;
        }

        bool mydone = false;
        for (;;) {
            if (tid == 0) s_min = 0x7FFFFFFF;
            __syncthreads();                       // A
            uint32_t pot[8]; uint32_t nz = 0u;
            if (!mydone) {
                #pragma unroll
                for (int w = 0; w < 8; ++w) { pot[w] = m[w] & ~D[w]; nz |= pot[w]; }
            }
            if (nz) atomicMin(&s_min, tid);
            __syncthreads();                       // B
            const int wmin = s_min;
            if (wmin == 0x7FFFFFFF) break;         // uniform exit
            if (tid == wmin) {
                // Earliest flagged proposal: evaluated under exactly its
                // sequential-time mask. Take first set bit (lowest label).
                int j = 0;
                #pragma unroll
                for (int w = 7; w >= 0; --w)
                    if (pot[w]) j = w * 32 + (__ffs(pot[w]) - 1);
                s_j = j;
                const int n = s_ntp;
                tp_idx[n]  = i;
                tp_conf[n] = conf[i];
                s_ntp = n + 1;
                mydone = true;
            }
            __syncthreads();                       // C
            const int j = s_j;
            D[j >> 5] |= (1u << (j & 31));         // all threads apply update
        }
    }
    __syncthreads();
    if (tid == 0) n_tp[0] = s_ntp;
}

// ---------------------------------------------------------------------------
// Kernel 3: stable-sort rank of each TP confidence (block per TP)
// ---------------------------------------------------------------------------
__global__ __launch_bounds__(256) void k_ranks(const float* __restrict__ conf,
                                               const int* __restrict__ tp_idx,
                                               const float* __restrict__ tp_conf,
                                               const int* __restrict__ n_tp,
                                               uint32_t* __restrict__ ranks) {
    const int m  = blockIdx.x;
    const int nt = *n_tp;
    if (m >= nt) return;                 // uniform per block, before barriers

    const float cg = tp_conf[m];
    const int   ig = tp_idx[m];
    const int   tid = threadIdx.x;

    int cnt = 0;
    for (int i = tid; i < N_PROPOSALS; i += 256) {
        const float c = conf[i];
        cnt += (c > cg) ? 1 : 0;
        cnt += (c == cg && i < ig) ? 1 : 0;   // stable tie-break by index
    }
    __shared__ int sc;
    if (tid == 0) sc = 0;
    __syncthreads();
    atomicAdd(&sc, cnt);
    __syncthreads();
    if (tid == 0) ranks[m] = (uint32_t)(1 + sc);
}

// ---------------------------------------------------------------------------
// Kernel 4: AP = (1/256) * sum over TPs (conf-desc order) of suffix-max(m/rank)
// ---------------------------------------------------------------------------
__global__ __launch_bounds__(256) void k_ap(const int* __restrict__ tp_idx,
                                            const float* __restrict__ tp_conf,
                                            const uint32_t* __restrict__ ranks,
                                            const int* __restrict__ n_tp,
                                            float* __restrict__ out) {
    __shared__ float pr[256];
    const int nt = *n_tp;
    const int m  = threadIdx.x;
    if (m < nt) {
        const float cm = tp_conf[m];
        const int   im = tp_idx[m];
        int tr = 1;                                  // rank among TPs (cumTP)
        for (int k = 0; k < nt; ++k) {
            const float ck = tp_conf[k];
            const int   ik = tp_idx[k];
            if (ck > cm || (ck == cm && ik < im)) ++tr;
        }
        pr[tr - 1] = (float)tr / (float)ranks[m];    // precision at this TP
    }
    __syncthreads();
    if (m == 0) {
        float P = 0.f, ap = 0.f;
        for (int k = nt - 1; k >= 0; --k) {          // suffix max (interp. AP)
            P = fmaxf(P, pr[k]);
            ap += P;
        }
        out[0] = ap / (float)N_LABELS;
    }
}

// ---------------------------------------------------------------------------
extern "C" void kernel_launch(void* const* d_in, const int* in_sizes, int n_in,
                              void* d_out, int out_size, void* d_ws, size_t ws_size,
                              hipStream_t stream) {
    (void)in_sizes; (void)n_in; (void)out_size; (void)ws_size;
    const float* props  = (const float*)d_in[0];   // (100000, 3) f32
    const float* labels = (const float*)d_in[1];   // (256, 2)   f32
    float*       out    = (float*)d_out;

    char*     ws    = (char*)d_ws;
    uint32_t* masks = (uint32_t*)(ws + OFF_MASK);
    float*    conf  = (float*)(ws + OFF_CONF);
    uint32_t* cor   = (uint32_t*)(ws + OFF_COR);
    int*      tpi   = (int*)(ws + OFF_TPIDX);
    float*    tpc   = (float*)(ws + OFF_TPCONF);
    int*      ntp   = (int*)(ws + OFF_NTP);
    uint32_t* rks   = (uint32_t*)(ws + OFF_RANKS);

    k_masks <<<NCHUNK, 1024, 0, stream>>>(props, labels, masks, conf, cor);
    k_greedy<<<1,      1024, 0, stream>>>(masks, cor, conf, tpi, tpc, ntp);
    k_ranks <<<256,     256, 0, stream>>>(conf, tpi, tpc, ntp, rks);
    k_ap    <<<1,       256, 0, stream>>>(tpi, tpc, rks, ntp, out);
}